// network_23089744183546
// MI455X (gfx1250) — compile-verified
//
#include <hip/hip_runtime.h>
#include <hip/hip_bf16.h>

#define BB 32
#define PP 1024
#define KNN 20
#define NPTS (BB*PP)
#define EPSV 1e-5f

typedef float v2f __attribute__((ext_vector_type(2)));
typedef float v8f __attribute__((ext_vector_type(8)));

// ---------- CDNA5 async global->LDS staging (ASYNCcnt-tracked) ----------
__device__ __forceinline__ void async_b128(void* lds, const void* g) {
    uint32_t off = (uint32_t)(uintptr_t)lds;   // generic addr low 32 bits = LDS offset
    asm volatile("global_load_async_to_lds_b128 %0, %1, off"
                 :: "v"(off), "v"(g) : "memory");
}
__device__ __forceinline__ void wait_async() {
    asm volatile("s_wait_asynccnt 0x0" ::: "memory");
}

__device__ __forceinline__ void topk_insert(float* bd, int* bi, float d, int j) {
    if (d >= bd[KNN-1]) return;
    int pos = KNN-1;
    while (pos > 0 && bd[pos-1] > d) { bd[pos] = bd[pos-1]; bi[pos] = bi[pos-1]; --pos; }
    bd[pos] = d; bi[pos] = j;
}

// ---------- K1: kNN on 3-d positions (cloud async-staged in LDS) ----------
__global__ void knn1_kernel(const float* __restrict__ pos, int* __restrict__ idx1) {
    __shared__ float sp[PP*3];
    int b = blockIdx.x >> 2;               // 4 blocks per cloud
    int base = b * PP;
    for (int c = threadIdx.x; c < PP*3/4; c += blockDim.x)     // 768 x b128 chunks
        async_b128(&sp[c*4], pos + base*3 + c*4);
    wait_async();
    __syncthreads();
    int p = (blockIdx.x & 3)*256 + threadIdx.x;
    float x = sp[p*3+0], y = sp[p*3+1], z = sp[p*3+2];
    float bd[KNN]; int bi[KNN];
    for (int k = 0; k < KNN; k++) { bd[k] = 3.4e38f; bi[k] = base; }
    for (int q = 0; q < PP; q++) {
        float dx = x - sp[q*3+0], dy = y - sp[q*3+1], dz = z - sp[q*3+2];
        topk_insert(bd, bi, dx*dx + dy*dy + dz*dz, base + q);  // includes self (d=0)
    }
    for (int k = 0; k < KNN; k++) idx1[(base+p)*KNN + k] = bi[k];
}

// ---------- K2: a0_i = x_i@(A1-B1)+b1a ; v0_i = x_i@B1  (W1a split) ----------
__global__ void av0_kernel(const float* __restrict__ pos, const float* __restrict__ W1a,
                           const float* __restrict__ b1a, float* __restrict__ a0,
                           float* __restrict__ v0) {
    int gid = blockIdx.x*blockDim.x + threadIdx.x;  // NPTS*64
    int i = gid >> 6, c = gid & 63;
    float av = b1a[c], vv = 0.f;
    for (int d = 0; d < 3; d++) {
        float xd = pos[i*3 + d];
        float wb = W1a[(3+d)*64 + c];
        av = fmaf(xd, W1a[d*64 + c] - wb, av);
        vv = fmaf(xd, wb, vv);
    }
    a0[gid] = av; v0[gid] = vv;
}

// ---------- K3/K4: BN stats over edges of h = a0_i + v0_j (deterministic 2-stage) ----------
__global__ void bn_partial_kernel(const float* __restrict__ a0, const float* __restrict__ v0,
                                  const int* __restrict__ idx1, float* __restrict__ partial) {
    __shared__ float rs[256], rq[256];
    int c = threadIdx.x & 63, sl = threadIdx.x >> 6;
    int i0 = blockIdx.x * 128;
    float s = 0.f, q = 0.f;
    for (int ii = sl; ii < 128; ii += 4) {
        int i = i0 + ii;
        float a = a0[i*64 + c];
        for (int k = 0; k < KNN; k++) {
            int j = idx1[i*KNN + k];
            float h = a + v0[j*64 + c];
            s += h; q = fmaf(h, h, q);
        }
    }
    rs[threadIdx.x] = s; rq[threadIdx.x] = q;
    __syncthreads();
    if (threadIdx.x < 64) {
        float S = 0.f, Q = 0.f;
        for (int t = 0; t < 4; t++) { S += rs[t*64 + c]; Q += rq[t*64 + c]; }
        partial[blockIdx.x*128 + c]      = S;
        partial[blockIdx.x*128 + 64 + c] = Q;
    }
}

__global__ void bn_final_kernel(const float* __restrict__ partial, const float* __restrict__ g1,
                                const float* __restrict__ be1, float* __restrict__ st) {
    int c = threadIdx.x;
    float S = 0.f, Q = 0.f;
    for (int b = 0; b < 256; b++) { S += partial[b*128 + c]; Q += partial[b*128 + 64 + c]; }
    const float N = (float)NPTS * (float)KNN;
    float mu  = S / N;
    float var = Q / N - mu*mu;
    float sc  = g1[c] * rsqrtf(var + EPSV);
    st[c]      = sc;                 // scale
    st[64 + c] = be1[c] - mu * sc;   // shift
}

// ---------- K5: EdgeConv1 via WMMA: H[20,64] in LDS, H@W1b, max over edges ----------
__global__ void conv1_kernel(const float* __restrict__ a0, const float* __restrict__ v0,
                             const int* __restrict__ idx1, const float* __restrict__ st,
                             const float* __restrict__ W1b, const float* __restrict__ b1b,
                             float* __restrict__ x1) {
    __shared__ float sW[64*72];      // W1b, row stride 72 (16B aligned, low-conflict)
    __shared__ float sH[4][32*68];   // per-wave edge features, rows 20..31 are don't-care
    for (int c = threadIdx.x; c < 1024; c += blockDim.x) {     // 64x64 = 1024 b128 chunks
        int r = c >> 4, j = (c & 15) * 4;
        async_b128(&sW[r*72 + j], W1b + r*64 + j);
    }
    int w = threadIdx.x >> 5, lane = threadIdx.x & 31;
    int i = blockIdx.x*4 + w;                       // one wave per point
    int c0 = lane, c1 = lane + 32;
    float s0 = st[c0], t0 = st[64+c0], s1 = st[c1], t1 = st[64+c1];
    float A0 = a0[i*64 + c0], A1 = a0[i*64 + c1];
    for (int k = 0; k < KNN; k++) {
        int j = idx1[i*KNN + k];
        float h0 = fmaf(s0, A0 + v0[j*64 + c0], t0); h0 = fmaxf(h0, 0.f);
        float h1 = fmaf(s1, A1 + v0[j*64 + c1], t1); h1 = fmaxf(h1, 0.f);
        sH[w][k*68 + c0] = h0; sH[w][k*68 + c1] = h1;
    }
    wait_async();
    __syncthreads();
    int l = lane & 15, hi = (lane >> 4) & 1;
    for (int nt = 0; nt < 4; nt++) {
        int n0 = nt*16;
        v8f cLo = {}, cHi = {};
        for (int k0 = 0; k0 < 64; k0 += 4) {
            v2f b, aL, aH;
            b.x  = sW[(k0 + 2*hi)*72 + n0 + l];
            b.y  = sW[(k0 + 2*hi + 1)*72 + n0 + l];
            aL.x = sH[w][l*68 + k0 + 2*hi];            // A rows = edges 0..15
            aL.y = sH[w][l*68 + k0 + 2*hi + 1];
            aH.x = sH[w][(16 + l)*68 + k0 + 2*hi];     // A rows = edges 16..31 (20..31 junk)
            aH.y = sH[w][(16 + l)*68 + k0 + 2*hi + 1];
            cLo = __builtin_amdgcn_wmma_f32_16x16x4_f32(false, aL, false, b, (short)0, cLo, false, false);
            cHi = __builtin_amdgcn_wmma_f32_16x16x4_f32(false, aH, false, b, (short)0, cHi, false, false);
        }
        float m = -3.4e38f;
        for (int r = 0; r < 8; r++) m = fmaxf(m, cLo[r]);       // edges r+8*hi (all valid)
        if (hi == 0)
            for (int r = 0; r < 4; r++) m = fmaxf(m, cHi[r]);   // edges 16..19 only
        float o = fmaxf(m, __shfl_xor(m, 16, 32));              // merge M-halves (same col)
        if (lane < 16) x1[i*64 + n0 + lane] = o + b1b[n0 + lane];
    }
}

// ---------- K6a: Wuv[64,256] = [ W2a - W2b | W2b ] ----------
__global__ void wuv_kernel(const float* __restrict__ W2, float* __restrict__ Wuv) {
    int gid = blockIdx.x*blockDim.x + threadIdx.x;  // 64*256
    int r = gid >> 8, c = gid & 255;
    float v = (c < 128) ? (W2[r*128 + c] - W2[(64+r)*128 + c]) : W2[(64+r)*128 + (c-128)];
    Wuv[r*256 + c] = v;
}

// ---------- K6b: uv[32768,256] = X1 @ Wuv (+b2 on U half)  -- f32 WMMA ----------
__global__ void uv_gemm_kernel(const float* __restrict__ x1, const float* __restrict__ Wuv,
                               const float* __restrict__ b2, float* __restrict__ uv) {
    __shared__ float As[16*76];
    int m0 = blockIdx.x * 16;
    {
        int t = threadIdx.x;            // 256 b128 chunks, one per thread
        int r = t >> 4, j = (t & 15) * 4;
        async_b128(&As[r*76 + j], x1 + (m0 + r)*64 + j);
    }
    wait_async();
    __syncthreads();
    int w = threadIdx.x >> 5, lane = threadIdx.x & 31;
    int l = lane & 15, hi = (lane >> 4) & 1;
    for (int sub = 0; sub < 2; sub++) {
        int n0 = w*32 + sub*16;
        v8f c = {};
        for (int k0 = 0; k0 < 64; k0 += 4) {
            v2f a, b;
            a.x = As[l*76 + k0 + 2*hi];
            a.y = As[l*76 + k0 + 2*hi + 1];
            b.x = Wuv[(k0 + 2*hi)*256 + n0 + l];
            b.y = Wuv[(k0 + 2*hi + 1)*256 + n0 + l];
            c = __builtin_amdgcn_wmma_f32_16x16x4_f32(false, a, false, b, (short)0, c, false, false);
        }
        int col = n0 + l;
        float bias = (col < 128) ? b2[col] : 0.f;
        for (int r = 0; r < 8; r++) {
            int row = m0 + r + 8*hi;   // C layout: vgpr r -> M = r + 8*hi, N = l
            uv[row*256 + col] = c[r] + bias;
        }
    }
}

// ---------- K7a: squared norms of x1 rows ----------
__global__ void d2_kernel(const float* __restrict__ x1, float* __restrict__ d2) {
    int i = blockIdx.x*blockDim.x + threadIdx.x;
    float s = 0.f;
    for (int k = 0; k < 64; k++) { float v = x1[i*64 + k]; s = fmaf(v, v, s); }
    d2[i] = s;
}

// ---------- K7b: kNN in 64-d via WMMA Gram tiles + fused per-lane top-20 ----------
__global__ void knn2_kernel(const float* __restrict__ x1, const float* __restrict__ d2,
                            int* __restrict__ idx2) {
    __shared__ float Ar[8][16*76];
    __shared__ float Qs[16*76];
    __shared__ float Dt[8][16*17];
    int b = blockIdx.x >> 3;
    int w = threadIdx.x >> 5, lane = threadIdx.x & 31;
    int l = lane & 15, hi = (lane >> 4) & 1;
    int p0 = (blockIdx.x & 7)*128 + w*16;     // row block within cloud (wave-owned)
    int rowg0 = b*PP + p0;
    for (int c = lane; c < 256; c += 32) {    // wave async-stages its own A rows
        int r = c >> 4, j = (c & 15) * 4;
        async_b128(&Ar[w][r*76 + j], x1 + (rowg0 + r)*64 + j);
    }
    float pd2[8];
    for (int r = 0; r < 8; r++) pd2[r] = d2[rowg0 + r + 8*hi];
    float bd[KNN]; int bi[KNN];
    for (int k = 0; k < KNN; k++) { bd[k] = 3.4e38f; bi[k] = b*PP; }
    for (int q0 = 0; q0 < PP; q0 += 16) {
        __syncthreads();
        {
            int t = threadIdx.x;              // stage q-tile: 256 b128 chunks
            int r = t >> 4, j = (t & 15) * 4;
            async_b128(&Qs[r*76 + j], x1 + (b*PP + q0 + r)*64 + j);
        }
        wait_async();
        __syncthreads();
        v8f c = {};
        for (int k0 = 0; k0 < 64; k0 += 4) {
            v2f a, bv;
            a.x  = Ar[w][l*76 + k0 + 2*hi];
            a.y  = Ar[w][l*76 + k0 + 2*hi + 1];
            bv.x = Qs[l*76 + k0 + 2*hi];       // B[k,n] = X1[q0+n, k]
            bv.y = Qs[l*76 + k0 + 2*hi + 1];
            c = __builtin_amdgcn_wmma_f32_16x16x4_f32(false, a, false, bv, (short)0, c, false, false);
        }
        float qd2 = d2[b*PP + q0 + l];
        for (int r = 0; r < 8; r++) {
            int m = r + 8*hi;
            Dt[w][m*17 + l] = pd2[r] + qd2 - 2.f*c[r];
        }
        __syncthreads();
        if (lane < 16) {
            for (int n = 0; n < 16; n++)
                topk_insert(bd, bi, Dt[w][lane*17 + n], b*PP + q0 + n);
        }
    }
    if (lane < 16) {
        int prow = rowg0 + lane;
        for (int k = 0; k < KNN; k++) idx2[prow*KNN + k] = bi[k];
    }
}

// ---------- K8: x2_i = u_i + max_k v_{idx2[i,k]} ----------
__global__ void conv2_fin_kernel(const float* __restrict__ uv, const int* __restrict__ idx2,
                                 float* __restrict__ x2) {
    int gid = blockIdx.x*blockDim.x + threadIdx.x;  // NPTS*128
    int i = gid >> 7, c = gid & 127;
    float m = -3.4e38f;
    for (int k = 0; k < KNN; k++) {
        int j = idx2[i*KNN + k];
        m = fmaxf(m, uv[j*256 + 128 + c]);
    }
    x2[gid] = uv[i*256 + c] + m;
}

// ---------- K9: [x1|x2] @ Wl fused with per-cloud max-pool  -- f32 WMMA ----------
__global__ void final_kernel(const float* __restrict__ x1, const float* __restrict__ x2,
                             const float* __restrict__ Wl, const float* __restrict__ bl,
                             float* __restrict__ out) {
    __shared__ float As[16*204];
    __shared__ float sm[8*32];
    int b = blockIdx.x;
    int w = threadIdx.x >> 5, lane = threadIdx.x & 31;
    int l = lane & 15, hi = (lane >> 4) & 1;
    int n0 = w*16;                               // wave owns 16 output channels
    float lmax = -3.4e38f;
    for (int m0 = 0; m0 < PP; m0 += 16) {
        __syncthreads();
        for (int c = threadIdx.x; c < 768; c += blockDim.x) {   // 16 rows x 48 chunks
            int r = c / 48, j = (c % 48) * 4;
            int row = b*PP + m0 + r;
            const float* src = (j < 64) ? (x1 + row*64 + j) : (x2 + row*128 + (j - 64));
            async_b128(&As[r*204 + j], src);
        }
        wait_async();
        __syncthreads();
        v8f c = {};
        for (int k0 = 0; k0 < 192; k0 += 4) {
            v2f a, bv;
            a.x  = As[l*204 + k0 + 2*hi];
            a.y  = As[l*204 + k0 + 2*hi + 1];
            bv.x = Wl[(k0 + 2*hi)*128 + n0 + l];
            bv.y = Wl[(k0 + 2*hi + 1)*128 + n0 + l];
            c = __builtin_amdgcn_wmma_f32_16x16x4_f32(false, a, false, bv, (short)0, c, false, false);
        }
        for (int r = 0; r < 8; r++) lmax = fmaxf(lmax, c[r]);  // max over this lane's 8 rows
    }
    sm[w*32 + lane] = lmax;
    __syncthreads();
    if (lane < 16) {
        float m = fmaxf(sm[w*32 + lane], sm[w*32 + 16 + lane]); // combine M halves (same col)
        out[b*128 + n0 + lane] = m + bl[n0 + lane];
    }
}

extern "C" void kernel_launch(void* const* d_in, const int* in_sizes, int n_in,
                              void* d_out, int out_size, void* d_ws, size_t ws_size,
                              hipStream_t stream) {
    const float* pos = (const float*)d_in[0];
    const float* W1a = (const float*)d_in[1];
    const float* b1a = (const float*)d_in[2];
    const float* g1  = (const float*)d_in[3];
    const float* be1 = (const float*)d_in[4];
    const float* W1b = (const float*)d_in[5];
    const float* b1b = (const float*)d_in[6];
    const float* W2  = (const float*)d_in[7];
    const float* b2  = (const float*)d_in[8];
    const float* Wl  = (const float*)d_in[9];
    const float* bl  = (const float*)d_in[10];
    float* out = (float*)d_out;

    char* base = (char*)d_ws;
    size_t off = 0;
    auto alloc = [&](size_t bytes) -> void* {
        void* r = base + off;
        off = (off + bytes + 255) & ~(size_t)255;
        return r;
    };
    int*   idx1 = (int*)  alloc((size_t)NPTS*KNN*4);   // kNN1 indices (global rows)
    float* a0   = (float*)alloc((size_t)NPTS*64*4);
    float* v0   = (float*)alloc((size_t)NPTS*64*4);
    float* part = (float*)alloc((size_t)256*128*4);
    float* st   = (float*)alloc((size_t)128*4);        // folded BN scale/shift
    float* x1   = (float*)alloc((size_t)NPTS*64*4);
    float* wuv  = (float*)alloc((size_t)64*256*4);
    float* uv   = (float*)alloc((size_t)NPTS*256*4);   // [U | V]
    float* d2v  = (float*)alloc((size_t)NPTS*4);
    int*   idx2 = (int*)  alloc((size_t)NPTS*KNN*4);
    float* x2   = (float*)alloc((size_t)NPTS*128*4);
    (void)ws_size; (void)in_sizes; (void)n_in; (void)out_size;

    knn1_kernel      <<<128, 256, 0, stream>>>(pos, idx1);
    av0_kernel       <<<NPTS*64/256, 256, 0, stream>>>(pos, W1a, b1a, a0, v0);
    bn_partial_kernel<<<256, 256, 0, stream>>>(a0, v0, idx1, part);
    bn_final_kernel  <<<1, 64, 0, stream>>>(part, g1, be1, st);
    conv1_kernel     <<<NPTS/4, 128, 0, stream>>>(a0, v0, idx1, st, W1b, b1b, x1);
    wuv_kernel       <<<64, 256, 0, stream>>>(W2, wuv);
    uv_gemm_kernel   <<<NPTS/16, 256, 0, stream>>>(x1, wuv, b2, uv);
    d2_kernel        <<<NPTS/256, 256, 0, stream>>>(x1, d2v);
    knn2_kernel      <<<256, 256, 0, stream>>>(x1, d2v, idx2);
    conv2_fin_kernel <<<NPTS*128/256, 256, 0, stream>>>(uv, idx2, x2);
    final_kernel     <<<BB, 256, 0, stream>>>(x1, x2, Wl, bl, out);
}